// GINStack_2173253452173
// MI455X (gfx1250) — compile-verified
//
#include <hip/hip_runtime.h>
#include <hip/hip_bf16.h>
#include <cstdint>

typedef float v8f __attribute__((ext_vector_type(8)));
typedef float v2f __attribute__((ext_vector_type(2)));

#define NN   8192
#define EE   131072
#define DD   768
#define DDE  876

// ---------------- GEMM tile config ----------------
#define BM    128
#define BN    128
#define KT    16
#define LDA_S 20    // KT + 4 pad  (80B rows: 16B aligned; frag b64 reads conflict-free)
#define LDB_S 136   // BN + 8 pad  (544B rows: 16B aligned; k / k+2 rows in disjoint bank windows)

// Epilogue / A-transform modes
//  0: C = acc + bias[col] + src[row*Nd+col]            (edge linear + h_src)
//  1: C = acc                                          (z, pre-batchnorm)
//  2: A-load transform relu(a*scale[k]+shift[k]); C stored at ldc/colOff (concat)
//  3: C = relu(acc + bias[col]) * dropout[row*Nd+col]
//  4: C = acc + bias[col]
template <int MODE>
__global__ __launch_bounds__(256)
void gemm_wmma_f32(const float* __restrict__ A, const float* __restrict__ B,
                   float* __restrict__ C, int M, int Nd, int K, int ldc, int colOff,
                   const float* __restrict__ bias, const float* __restrict__ src,
                   const float* __restrict__ scale, const float* __restrict__ shift,
                   const float* __restrict__ dropout)
{
    __shared__ float As[2][BM * LDA_S];
    __shared__ float Bs[2][KT * LDB_S];

    const int tid  = threadIdx.x;
    const int lane = tid & 31;
    const int wave = tid >> 5;
    const int half = lane >> 4;
    const int l16  = lane & 15;
    const int wrow = wave >> 1;   // 0..3 -> 32-row strips
    const int wcol = wave & 1;    // 0..1 -> 64-col strips
    const int blockM = blockIdx.y * BM;
    const int blockN = blockIdx.x * BN;

    v8f acc[2][4] = {{{}, {}, {}, {}}, {{}, {}, {}, {}}};

    // ---- staging helpers ----
    auto stageA_fast = [&](float* dst, int k0) {
        #pragma unroll
        for (int r = 0; r < 2; ++r) {
            int f4  = tid + r * 256;        // 0..511
            int row = f4 >> 2;              // 0..127
            int kc  = (f4 & 3) << 2;        // 0,4,8,12
            int gk  = k0 + kc;
            const float* ap = A + (size_t)(blockM + row) * K + gk;
            float4 v = *(const float4*)ap;
            __builtin_prefetch(ap + 2 * KT, 0, 0);      // global_prefetch_b8
            if (MODE == 2) {                            // fused batchnorm + relu on A path
                v.x = fmaxf(v.x * scale[gk + 0] + shift[gk + 0], 0.f);
                v.y = fmaxf(v.y * scale[gk + 1] + shift[gk + 1], 0.f);
                v.z = fmaxf(v.z * scale[gk + 2] + shift[gk + 2], 0.f);
                v.w = fmaxf(v.w * scale[gk + 3] + shift[gk + 3], 0.f);
            }
            *(float4*)&dst[row * LDA_S + kc] = v;
        }
    };
    auto stageA_guard = [&](float* dst, int k0) {
        #pragma unroll
        for (int r = 0; r < 2; ++r) {
            int f4  = tid + r * 256;
            int row = f4 >> 2;
            int kc  = (f4 & 3) << 2;
            int gk  = k0 + kc;
            const float* ap = A + (size_t)(blockM + row) * K + gk;
            float4 v;
            v.x = (gk + 0 < K) ? ap[0] : 0.f;
            v.y = (gk + 1 < K) ? ap[1] : 0.f;
            v.z = (gk + 2 < K) ? ap[2] : 0.f;
            v.w = (gk + 3 < K) ? ap[3] : 0.f;
            if (MODE == 2) {
                v.x = (gk + 0 < K) ? fmaxf(v.x * scale[gk + 0] + shift[gk + 0], 0.f) : 0.f;
                v.y = (gk + 1 < K) ? fmaxf(v.y * scale[gk + 1] + shift[gk + 1], 0.f) : 0.f;
                v.z = (gk + 2 < K) ? fmaxf(v.z * scale[gk + 2] + shift[gk + 2], 0.f) : 0.f;
                v.w = (gk + 3 < K) ? fmaxf(v.w * scale[gk + 3] + shift[gk + 3], 0.f) : 0.f;
            }
            *(float4*)&dst[row * LDA_S + kc] = v;
        }
    };
    auto stageB = [&](float* dst, int k0, bool guard) {
        #pragma unroll
        for (int r = 0; r < 2; ++r) {
            int idx = tid + r * 256;        // 0..511
            int kk  = idx >> 5;             // 0..15 (32 float4 per row)
            int nc  = (idx & 31) << 2;      // 0..124
            int gk  = k0 + kk;
            float4 v = make_float4(0.f, 0.f, 0.f, 0.f);
            if (!guard || gk < K)
                v = *(const float4*)(B + (size_t)gk * Nd + blockN + nc);
            *(float4*)&dst[kk * LDB_S + nc] = v;
        }
    };
    auto compute = [&](const float* As_, const float* Bs_) {
        #pragma unroll
        for (int ks = 0; ks < KT; ks += 4) {
            v2f af[2], bf[4];
            #pragma unroll
            for (int mt = 0; mt < 2; ++mt) {
                // A frag 16x4: lanes 0-15 K={0,1}, lanes 16-31 K={2,3}
                const float* p = &As_[(wrow * 32 + mt * 16 + l16) * LDA_S + ks + 2 * half];
                af[mt].x = p[0];
                af[mt].y = p[1];
            }
            #pragma unroll
            for (int nt = 0; nt < 4; ++nt) {
                int col = wcol * 64 + nt * 16 + l16;
                bf[nt].x = Bs_[(ks + 2 * half + 0) * LDB_S + col];
                bf[nt].y = Bs_[(ks + 2 * half + 1) * LDB_S + col];
            }
            #pragma unroll
            for (int mt = 0; mt < 2; ++mt)
                #pragma unroll
                for (int nt = 0; nt < 4; ++nt)
                    acc[mt][nt] = __builtin_amdgcn_wmma_f32_16x16x4_f32(
                        false, af[mt], false, bf[nt], (short)0, acc[mt][nt], false, false);
        }
    };

    // ---- K loop: double-buffered LDS, unguarded full chunks + one guarded tail ----
    const int kFull = K / KT;           // #full chunks (K >= KT always here)
    const int rem   = K - kFull * KT;

    stageA_fast(As[0], 0);
    stageB(Bs[0], 0, false);
    __syncthreads();

    int cur = 0;
    for (int c = 0; c < kFull; ++c) {
        const int knext = (c + 1) * KT;
        if (c + 1 < kFull) {
            stageA_fast(As[cur ^ 1], knext);
            stageB(Bs[cur ^ 1], knext, false);
        } else if (rem > 0) {
            stageA_guard(As[cur ^ 1], knext);
            stageB(Bs[cur ^ 1], knext, true);
        }
        compute(As[cur], Bs[cur]);
        __syncthreads();
        cur ^= 1;
    }
    if (rem > 0) compute(As[cur], Bs[cur]);

    // ---- epilogue: C/D layout => VGPR j holds M = j + 8*half, N = l16 ----
    #pragma unroll
    for (int mt = 0; mt < 2; ++mt) {
        #pragma unroll
        for (int nt = 0; nt < 4; ++nt) {
            int gcol = blockN + wcol * 64 + nt * 16 + l16;
            #pragma unroll
            for (int j = 0; j < 8; ++j) {
                int grow = blockM + wrow * 32 + mt * 16 + j + 8 * half;
                float v = acc[mt][nt][j];
                if (MODE == 0) v += bias[gcol] + src[(size_t)grow * Nd + gcol];
                if (MODE == 3) v = fmaxf(v + bias[gcol], 0.f) * dropout[(size_t)grow * Nd + gcol];
                if (MODE == 4) v += bias[gcol];
                C[(size_t)grow * ldc + colOff + gcol] = v;
            }
        }
    }
}

// ---------------- edge scatter: msgs[seg[e]] += relu(edge_feats[e]) ----------------
__global__ __launch_bounds__(256)
void edge_scatter(const float* __restrict__ ef, const int* __restrict__ seg,
                  float* __restrict__ msgs)
{
    const int e = blockIdx.x;
    const int c = threadIdx.x;                 // DE/4 = 219 float4 lanes
    if (c >= DDE / 4) return;
    const int s = seg[e];
    float4 v = ((const float4*)(ef + (size_t)e * DDE))[c];
    float* dst = msgs + (size_t)s * DDE + c * 4;
    atomicAdd(dst + 0, fmaxf(v.x, 0.f));
    atomicAdd(dst + 1, fmaxf(v.y, 0.f));
    atomicAdd(dst + 2, fmaxf(v.z, 0.f));
    atomicAdd(dst + 3, fmaxf(v.w, 0.f));
}

// ---------------- tiled transpose: W[Dout][K] -> WT[K][Dout] ----------------
__global__ __launch_bounds__(256)
void transpose_w(const float* __restrict__ W, float* __restrict__ WT, int Dout, int K)
{
    __shared__ float tile[32][33];
    const int kb = blockIdx.x * 32, nb = blockIdx.y * 32;
    const int tx = threadIdx.x & 31, ty = threadIdx.x >> 5;
    for (int r = ty; r < 32; r += 8) {
        int n = nb + r, k = kb + tx;
        tile[r][tx] = (n < Dout && k < K) ? W[(size_t)n * K + k] : 0.f;
    }
    __syncthreads();
    for (int r = ty; r < 32; r += 8) {
        int k = kb + r, n = nb + tx;
        if (k < K && n < Dout) WT[(size_t)k * Dout + n] = tile[tx][r];
    }
}

// ---------------- batchnorm column statistics ----------------
__global__ __launch_bounds__(256)
void colstats(const float* __restrict__ z, float* __restrict__ sum,
              float* __restrict__ sumsq, int Nrows)
{
    const int d    = blockIdx.x * 256 + threadIdx.x;  // column
    const int rows = Nrows / gridDim.y;
    const int r0   = blockIdx.y * rows;
    float s = 0.f, ss = 0.f;
    for (int r = 0; r < rows; ++r) {
        float v = z[(size_t)(r0 + r) * DD + d];
        s += v; ss += v * v;
    }
    atomicAdd(&sum[d], s);
    atomicAdd(&sumsq[d], ss);
}

__global__ __launch_bounds__(256)
void bn_finalize(const float* __restrict__ sum, const float* __restrict__ sumsq,
                 const float* __restrict__ g, const float* __restrict__ b,
                 float* __restrict__ scale, float* __restrict__ shift, float invN)
{
    const int d = blockIdx.x * 256 + threadIdx.x;
    if (d >= DD) return;
    float mu  = sum[d] * invN;
    float var = sumsq[d] * invN - mu * mu;
    float is  = rsqrtf(var + 1e-5f);
    float sc  = g[d] * is;
    scale[d]  = sc;
    shift[d]  = b[d] - mu * sc;
}

extern "C" void kernel_launch(void* const* d_in, const int* in_sizes, int n_in,
                              void* d_out, int out_size, void* d_ws, size_t ws_size,
                              hipStream_t stream)
{
    (void)in_sizes; (void)n_in; (void)out_size; (void)ws_size;

    const float* paths = (const float*)d_in[0];
    const float* ef    = (const float*)d_in[1];
    const int*   seg   = (const int*)  d_in[2];
    const float* drop  = (const float*)d_in[3];
    const float *We[3], *be[3], *Wa[3], *Wb[3], *gn[3], *bnb[3];
    for (int i = 0; i < 3; ++i) {
        int b = 4 + 6 * i;
        We[i]  = (const float*)d_in[b + 0];
        be[i]  = (const float*)d_in[b + 1];
        Wa[i]  = (const float*)d_in[b + 2];
        Wb[i]  = (const float*)d_in[b + 3];
        gn[i]  = (const float*)d_in[b + 4];
        bnb[i] = (const float*)d_in[b + 5];
    }
    const float* l1w = (const float*)d_in[22];
    const float* l1b = (const float*)d_in[23];
    const float* l2w = (const float*)d_in[24];
    const float* l2b = (const float*)d_in[25];
    float* out = (float*)d_out;

    // ---- workspace layout (floats) ----
    float* ws = (float*)d_ws;
    size_t o = 0;
    float* msgs  = ws + o; o += (size_t)NN * DDE;
    float* h     = ws + o; o += (size_t)NN * DD;
    float* z     = ws + o; o += (size_t)NN * DD;
    float* hcat  = ws + o; o += (size_t)NN * 3 * DD;
    float* hd    = ws + o; o += (size_t)NN * DD;
    float* WeT[3], *WaT[3], *WbT[3];
    for (int i = 0; i < 3; ++i) { WeT[i] = ws + o; o += (size_t)DDE * DD; }
    for (int i = 0; i < 3; ++i) { WaT[i] = ws + o; o += (size_t)DD * DD; }
    for (int i = 0; i < 3; ++i) { WbT[i] = ws + o; o += (size_t)DD * DD; }
    float* l1T = ws + o; o += (size_t)3 * DD * DD;
    float* l2T = ws + o; o += (size_t)DD * DD;
    float* csum  = ws + o; o += DD;
    float* csq   = ws + o; o += DD;
    float* scale = ws + o; o += DD;
    float* shift = ws + o; o += DD;

    const dim3 blk(256);
    const dim3 gemm_grid(DD / BN, NN / BM);          // 6 x 64

    // shared segment-sum of relu(edge_feats) (identical for all three convs)
    hipMemsetAsync(msgs, 0, (size_t)NN * DDE * sizeof(float), stream);
    edge_scatter<<<EE, blk, 0, stream>>>(ef, seg, msgs);

    // pre-transpose weights -> row-major [K][Dout] for coalesced B tiles
    for (int i = 0; i < 3; ++i) {
        transpose_w<<<dim3((DDE + 31) / 32, DD / 32), blk, 0, stream>>>(We[i], WeT[i], DD, DDE);
        transpose_w<<<dim3(DD / 32, DD / 32), blk, 0, stream>>>(Wa[i], WaT[i], DD, DD);
        transpose_w<<<dim3(DD / 32, DD / 32), blk, 0, stream>>>(Wb[i], WbT[i], DD, DD);
    }
    transpose_w<<<dim3(3 * DD / 32, DD / 32), blk, 0, stream>>>(l1w, l1T, DD, 3 * DD);
    transpose_w<<<dim3(DD / 32, DD / 32), blk, 0, stream>>>(l2w, l2T, DD, DD);

    for (int i = 0; i < 3; ++i) {
        // h = msgs @ We^T + be + h_src
        gemm_wmma_f32<0><<<gemm_grid, blk, 0, stream>>>(msgs, WeT[i], h, NN, DD, DDE,
                                                        DD, 0, be[i], paths, nullptr, nullptr, nullptr);
        // z = h @ Wa^T
        gemm_wmma_f32<1><<<gemm_grid, blk, 0, stream>>>(h, WaT[i], z, NN, DD, DD,
                                                        DD, 0, nullptr, nullptr, nullptr, nullptr, nullptr);
        // batchnorm training statistics
        hipMemsetAsync(csum, 0, 2 * DD * sizeof(float), stream);
        colstats<<<dim3(DD / 256, 64), blk, 0, stream>>>(z, csum, csq, NN);
        bn_finalize<<<dim3(DD / 256), blk, 0, stream>>>(csum, csq, gn[i], bnb[i],
                                                        scale, shift, 1.f / (float)NN);
        // hcat[:, i*D:(i+1)*D] = relu(bn(z)) @ Wb^T   (BN+relu fused on A path)
        gemm_wmma_f32<2><<<gemm_grid, blk, 0, stream>>>(z, WbT[i], hcat, NN, DD, DD,
                                                        3 * DD, i * DD, nullptr, nullptr,
                                                        scale, shift, nullptr);
    }

    // hd = relu(hcat @ lin1^T + b1) * dropout
    gemm_wmma_f32<3><<<gemm_grid, blk, 0, stream>>>(hcat, l1T, hd, NN, DD, 3 * DD,
                                                    DD, 0, l1b, nullptr, nullptr, nullptr, drop);
    // out = hd @ lin2^T + b2
    gemm_wmma_f32<4><<<gemm_grid, blk, 0, stream>>>(hd, l2T, out, NN, DD, DD,
                                                    DD, 0, l2b, nullptr, nullptr, nullptr, nullptr);
}